// GCN_33234456937223
// MI455X (gfx1250) — compile-verified
//
#include <hip/hip_runtime.h>
#include <cstddef>

// ---------------------------------------------------------------------------
// GCN on MI455X (gfx1250, wave32).
// Per layer, with g[i] = (h@W)[i]*dinv[i]:
//   h'[i] = tanh( dinv[i] * ( sum_{e:dst=i} g[src_e] + g[i] ) + b )
// Edge hot loop: pure gather + global_atomic_add_f32 scatter (zero FLOPs).
// Dense transforms: V_WMMA_F32_16X16X4_F32, D extracted via LDS to get one
// coalesced b128 store per node instead of 8 exec-masked scalar stores.
// ---------------------------------------------------------------------------

typedef __attribute__((ext_vector_type(2))) float v2f;
typedef __attribute__((ext_vector_type(8))) float v8f;

__global__ void zero_int_kernel(int* __restrict__ p, int n) {
  int i = blockIdx.x * blockDim.x + threadIdx.x;
  if (i < n) p[i] = 0;
}

__global__ void degree_kernel(const int* __restrict__ dst, int* __restrict__ cnt, int E) {
  int e = blockIdx.x * blockDim.x + threadIdx.x;
  if (e < E) atomicAdd(&cnt[dst[e]], 1);
}

// Layer 1 init: dinv = rsqrt(deg+1);  g1 = x * dinv * W1(1x4);  accA = 0
__global__ void init_layer1_kernel(const float* __restrict__ x,
                                   const int* __restrict__ cnt,
                                   const float* __restrict__ W1,
                                   float* __restrict__ dinv,
                                   float* __restrict__ g1,
                                   float* __restrict__ accA, int N) {
  int i = blockIdx.x * blockDim.x + threadIdx.x;
  if (i >= N) return;
  float di = rsqrtf((float)cnt[i] + 1.0f);
  dinv[i] = di;
  float xs = x[i] * di;
  float4 g;
  g.x = xs * W1[0]; g.y = xs * W1[1]; g.z = xs * W1[2]; g.w = xs * W1[3];
  reinterpret_cast<float4*>(g1)[i] = g;
  reinterpret_cast<float4*>(accA)[i] = make_float4(0.f, 0.f, 0.f, 0.f);
}

// Hot loop: acc[dst] += g[src].  Pure gather + f32 atomics.
template <int F>
__global__ void edge_scatter_kernel(const int* __restrict__ src,
                                    const int* __restrict__ dst,
                                    const float* __restrict__ g,
                                    float* __restrict__ acc, int E) {
  int e = blockIdx.x * blockDim.x + threadIdx.x;
  if (e >= E) return;
  int s = src[e];
  int d = dst[e];
  if (F == 4) {
    float4 v = reinterpret_cast<const float4*>(g)[s];
    float* a = acc + (size_t)d * 4;
    unsafeAtomicAdd(a + 0, v.x);
    unsafeAtomicAdd(a + 1, v.y);
    unsafeAtomicAdd(a + 2, v.z);
    unsafeAtomicAdd(a + 3, v.w);
  } else {
    float2 v = reinterpret_cast<const float2*>(g)[s];
    float* a = acc + (size_t)d * 2;
    unsafeAtomicAdd(a + 0, v.x);
    unsafeAtomicAdd(a + 1, v.y);
  }
}

// Fused: h = tanh(dinv*(acc_in+g_in)+b_prev); g_out = (h @ W[4xFOUT]) * dinv;
// acc_out = 0.  One 16-node tile per wave via V_WMMA_F32_16X16X4_F32.
// A (16x4 f32, 2 VGPRs): lanes 0-15 hold K=0,1; lanes 16-31 hold K=2,3 (ISA 7.12.2).
// B (4x16) mirrored: lane = N-col, vgpr0/1 = K rows {2*half, 2*half+1}.
// D (16x16): vgpr v = row v (lanes 0-15) / row v+8 (lanes 16-31), lane%16 = col.
// D extraction goes through a wave-private LDS slot so each node is written
// with a single coalesced float4/float2 store.
template <int FOUT>
__launch_bounds__(256)
__global__ void transform_wmma_kernel(const float* __restrict__ acc_in,
                                      const float* __restrict__ g_in,
                                      const float* __restrict__ dinv,
                                      const float* __restrict__ bprev,  // [4]
                                      const float* __restrict__ W,      // [4][FOUT] row-major
                                      float* __restrict__ g_out,
                                      float* __restrict__ acc_out, int N) {
  __shared__ float xchg[8 * 16 * FOUT];          // 8 waves/block, 16 nodes * FOUT
  const int tid  = blockIdx.x * blockDim.x + threadIdx.x;
  const int wave = tid >> 5;                     // uniform within wave
  const int base = wave * 16;
  if (base >= N) return;                         // wave-uniform exit
  const int lane = threadIdx.x & 31;
  const int half = lane >> 4;
  const int ln   = lane & 15;
  const int kb   = 2 * half;                     // this lane-half supplies K = kb, kb+1
  float* lds = &xchg[(threadIdx.x >> 5) * (16 * FOUT)];

  // ---- build A: 16 nodes x 4 features of tanh-activated previous layer ----
  int node = base + ln;
  const bool in_range = (node < N);
  if (!in_range) node = N - 1;                   // branchless clamp keeps EXEC full
  const float di = dinv[node];
  const size_t nb = (size_t)node * 4;
  float h0 = tanhf(di * (acc_in[nb + kb]     + g_in[nb + kb])     + bprev[kb]);
  float h1 = tanhf(di * (acc_in[nb + kb + 1] + g_in[nb + kb + 1]) + bprev[kb + 1]);
  v2f a; a.x = h0; a.y = h1;

  // ---- build B: W padded to 4x16 (columns >= FOUT are zero) ----
  int cidx = (ln < FOUT) ? ln : 0;
  float bx = W[(size_t)kb * FOUT + cidx];
  float by = W[(size_t)(kb + 1) * FOUT + cidx];
  if (ln >= FOUT) { bx = 0.f; by = 0.f; }
  v2f b; b.x = bx; b.y = by;

  v8f c = {};
  c = __builtin_amdgcn_wmma_f32_16x16x4_f32(
      /*neg_a=*/false, a, /*neg_b=*/false, b,
      /*c_mod=*/(short)0, c, /*reuse_a=*/false, /*reuse_b=*/false);

  // ---- scatter D into the wave's LDS slot (node-major layout) ----
  if (ln < FOUT) {
#pragma unroll
    for (int v = 0; v < 8; ++v)
      lds[(v + 8 * half) * FOUT + ln] = c[v];
  }
  __builtin_amdgcn_wave_barrier();               // wave-synchronous LDS exchange

  // ---- one coalesced vector store per node: g_out[node] = D[node,:]*dinv ----
  if (in_range) {                                // node == base + ln here
    if (FOUT == 4) {
      float4 gv;
      gv.x = lds[ln * 4 + 0] * di;
      gv.y = lds[ln * 4 + 1] * di;
      gv.z = lds[ln * 4 + 2] * di;
      gv.w = lds[ln * 4 + 3] * di;
      reinterpret_cast<float4*>(g_out)[node] = gv;
    } else {
      float2 gv;
      gv.x = lds[ln * 2 + 0] * di;
      gv.y = lds[ln * 2 + 1] * di;
      reinterpret_cast<float2*>(g_out)[node] = gv;
    }
  }

  // ---- zero this tile's slice of acc_out (16*FOUT floats per wave) ----
#pragma unroll
  for (int t = 0; t < FOUT / 2; ++t) {
    size_t idx = (size_t)base * FOUT + (size_t)t * 32 + lane;
    if (idx < (size_t)N * FOUT) acc_out[idx] = 0.f;
  }
}

// Final: h3 = tanh(dinv*(acc3+g3)+b3); out = h3 @ Wc + bc; emit [out | h3].
__global__ void finalize_kernel(const float* __restrict__ acc3,
                                const float* __restrict__ g3,
                                const float* __restrict__ dinv,
                                const float* __restrict__ b3,
                                const float* __restrict__ Wc,
                                const float* __restrict__ bc,
                                float* __restrict__ out, int N) {
  int i = blockIdx.x * blockDim.x + threadIdx.x;
  if (i >= N) return;
  float di = dinv[i];
  float2 av = reinterpret_cast<const float2*>(acc3)[i];
  float2 gv = reinterpret_cast<const float2*>(g3)[i];
  float h0 = tanhf(di * (av.x + gv.x) + b3[0]);
  float h1 = tanhf(di * (av.y + gv.y) + b3[1]);
  out[i] = h0 * Wc[0] + h1 * Wc[1] + bc[0];
  out[(size_t)N + 2 * i]     = h0;
  out[(size_t)N + 2 * i + 1] = h1;
}

extern "C" void kernel_launch(void* const* d_in, const int* in_sizes, int n_in,
                              void* d_out, int out_size, void* d_ws, size_t ws_size,
                              hipStream_t stream) {
  const float* x  = (const float*)d_in[0];
  const int*   ei = (const int*)d_in[1];
  const float* W1 = (const float*)d_in[2];
  const float* b1 = (const float*)d_in[3];
  const float* W2 = (const float*)d_in[4];
  const float* b2 = (const float*)d_in[5];
  const float* W3 = (const float*)d_in[6];
  const float* b3 = (const float*)d_in[7];
  const float* Wc = (const float*)d_in[8];
  const float* bc = (const float*)d_in[9];

  const int N = in_sizes[0];
  const int E = in_sizes[1] / 2;
  const int* src = ei;
  const int* dst = ei + E;

  // Workspace layout (floats), offsets padded to keep float4 alignment.
  size_t Np = ((size_t)N + 3) & ~(size_t)3;
  float* ws   = (float*)d_ws;
  int*   cnt  = (int*)ws;            // N ints
  float* dinv = ws + Np;             // N
  float* g1   = ws + 2 * Np;         // 4N
  float* accA = ws + 6 * Np;         // 4N
  float* g2   = ws + 10 * Np;        // 4N
  float* accB = ws + 14 * Np;        // 4N
  float* g3   = ws + 18 * Np;        // 2N
  float* acc3 = ws + 20 * Np;        // 2N  (total 22N floats ~ 44 MB)

  const int BLK = 256;
  const int gbN = (N + BLK - 1) / BLK;
  const int gbE = (E + BLK - 1) / BLK;
  const int waves = (N + 15) / 16;                 // one 16-node tile per wave
  const int gbW = (waves * 32 + BLK - 1) / BLK;

  float* out = (float*)d_out;

  // degree (integer atomics) -> dinv
  zero_int_kernel<<<gbN, BLK, 0, stream>>>(cnt, N);
  degree_kernel<<<gbE, BLK, 0, stream>>>(dst, cnt, E);

  // layer 1: 1->4 (scalar transform) + edge scatter
  init_layer1_kernel<<<gbN, BLK, 0, stream>>>(x, cnt, W1, dinv, g1, accA, N);
  edge_scatter_kernel<4><<<gbE, BLK, 0, stream>>>(src, dst, g1, accA, E);

  // layer 2: tanh+4x4 transform via WMMA, then edge scatter
  transform_wmma_kernel<4><<<gbW, BLK, 0, stream>>>(accA, g1, dinv, b1, W2, g2, accB, N);
  edge_scatter_kernel<4><<<gbE, BLK, 0, stream>>>(src, dst, g2, accB, E);

  // layer 3: tanh+4x2 transform via WMMA, then edge scatter
  transform_wmma_kernel<2><<<gbW, BLK, 0, stream>>>(accB, g2, dinv, b2, W3, g3, acc3, N);
  edge_scatter_kernel<2><<<gbE, BLK, 0, stream>>>(src, dst, g3, acc3, E);

  // classifier + outputs
  finalize_kernel<<<gbN, BLK, 0, stream>>>(acc3, g3, dinv, b3, Wc, bc, out, N);
}